// HierAttNet_40475771798139
// MI455X (gfx1250) — compile-verified
//
#include <hip/hip_runtime.h>
#include <hip/hip_bf16.h>

typedef __attribute__((ext_vector_type(2))) float v2f;
typedef __attribute__((ext_vector_type(8))) float v8f;

#define B_     8
#define ND_    512
#define D_     128
#define NV_    4096
#define NODES_ 512

// ---------------------------------------------------------------------------
// Kernel 1: fused  sim = emb[doc] @ VvT  ->  digitize -> bin_w gather ->
//           attn-weighted reduction over n  ->  a[b, v]   (a in d_ws)
// grid = (NV/128, B), block = 256 (8 waves). Wave w handles v-columns
// [blockIdx.x*128 + 16w, +16) for the full n = 0..511 range of batch b.
// ---------------------------------------------------------------------------
__global__ __launch_bounds__(256) void hier_sim_bin_attn_kernel(
    const int*   __restrict__ doc_index,   // [B, ND]
    const float* __restrict__ attn,        // [B, ND]
    const float* __restrict__ emb,         // [VOCAB, D]
    const float* __restrict__ VvT,         // [D, NV]
    const float* __restrict__ bin_diff,    // [16]
    const float* __restrict__ bin_start,   // [1]
    float*       __restrict__ a_out)       // [B, NV]
{
    __shared__ float attn_lds[ND_];
    __shared__ float bin_lds[16];

    const int b    = blockIdx.y;
    const int tid  = threadIdx.x;
    const int wave = tid >> 5;
    const int lane = tid & 31;
    const int col  = lane & 15;     // N within 16-wide tile
    const int hi   = lane >> 4;     // 0 -> K pair {0,1}, 1 -> {2,3} (f32 A/B frag)

    // stage attn row for this batch into LDS
    for (int i = tid; i < ND_; i += 256)
        attn_lds[i] = attn[b * ND_ + i];
    // bin_w = start + cumsum(relu(diff))  (16 entries, serial cumsum)
    if (tid == 0) {
        float run = bin_start[0];
        #pragma unroll
        for (int i = 0; i < 16; ++i) {
            float d = bin_diff[i];
            run += (d > 0.0f) ? d : 0.0f;
            bin_lds[i] = run;
        }
    }
    __syncthreads();

    const int v0 = blockIdx.x * 128 + wave * 16;

    // Hoist all B-fragments (VvT) for K = 0..127: 32 steps x v2f = 64 VGPRs.
    v2f Bf[32];
    #pragma unroll
    for (int s = 0; s < 32; ++s) {
        const int k = 4 * s + 2 * hi;
        Bf[s].x = VvT[(size_t)k       * NV_ + v0 + col];
        Bf[s].y = VvT[(size_t)(k + 1) * NV_ + v0 + col];
    }

    // digitize constants: midpoints = linspace(-0.5, 0.99, 15), step h
    const float inv_h = 14.0f / 1.49f;

    float acc = 0.0f;                          // partial a[b, v0+col] (per M-half)
    for (int nt = 0; nt < 32; ++nt) {          // 32 n-tiles of 16 rows
        const int nbase = nt * 16;
        const int row   = doc_index[b * ND_ + nbase + col];
        const float* arow = emb + (size_t)row * D_ + 2 * hi;

        v8f c = {};
        #pragma unroll
        for (int s = 0; s < 32; ++s) {
            v2f af = *(const v2f*)(arow + 4 * s);   // global_load_b64
            c = __builtin_amdgcn_wmma_f32_16x16x4_f32(
                    false, af, false, Bf[s], (short)0, c, false, false);
        }

        // C/D layout: reg r, lanes 0-15 -> (M=r, N=lane); lanes 16-31 -> (M=r+8)
        #pragma unroll
        for (int r = 0; r < 8; ++r) {
            const float x = c[r];
            const float t = (x + 0.5f) * inv_h;
            int dig = (t < 0.0f) ? 0 : (int)t + 1;
            dig = (dig > 15) ? 15 : dig;
            const float w = bin_lds[dig];
            const int   m = r + 8 * hi;
            acc += w * attn_lds[nbase + m];
        }
    }

    // combine the two M-halves (lane L + lane L+16), lanes 0..15 write out
    acc += __shfl_xor(acc, 16, 32);
    if (hi == 0)
        a_out[(size_t)b * NV_ + v0 + col] = acc;
}

// ---------------------------------------------------------------------------
// Kernel 2: final = a @ phi_vs^T   ([8,4096] x [4096,512] -> [8,512])
// M padded 8 -> 16 with zero rows; one wave per 16-wide node tile.
// ---------------------------------------------------------------------------
__global__ __launch_bounds__(32) void hier_final_gemm_kernel(
    const float* __restrict__ a,     // [8, NV]
    const float* __restrict__ phi,   // [NODES, NV]
    float*       __restrict__ out)   // [8, NODES]
{
    const int j0   = blockIdx.x * 16;
    const int lane = threadIdx.x & 31;
    const int col  = lane & 15;
    const int hi   = lane >> 4;
    const int m    = col;                       // A-matrix row for this lane

    const float* arow = a   + (size_t)m * NV_ + 2 * hi;          // valid if m < 8
    const float* prow = phi + (size_t)(j0 + col) * NV_ + 2 * hi; // B[k][n]=phi[j0+n][k]

    v8f c = {};
    for (int s = 0; s < NV_ / 4; ++s) {
        v2f af;
        if (m < 8) af = *(const v2f*)(arow + 4 * s);
        else       { af.x = 0.0f; af.y = 0.0f; }   // zero-pad rows 8..15
        v2f bf = *(const v2f*)(prow + 4 * s);
        c = __builtin_amdgcn_wmma_f32_16x16x4_f32(
                false, af, false, bf, (short)0, c, false, false);
    }

    #pragma unroll
    for (int r = 0; r < 8; ++r) {
        const int M = r + 8 * hi;
        if (M < 8)
            out[(size_t)M * NODES_ + j0 + col] = c[r];
    }
}

// ---------------------------------------------------------------------------
extern "C" void kernel_launch(void* const* d_in, const int* in_sizes, int n_in,
                              void* d_out, int out_size, void* d_ws, size_t ws_size,
                              hipStream_t stream) {
    const int*   doc_index = (const int*)  d_in[0];
    const float* attn      = (const float*)d_in[1];
    const float* emb       = (const float*)d_in[2];
    const float* VvT       = (const float*)d_in[3];
    const float* phi       = (const float*)d_in[4];
    const float* bin_diff  = (const float*)d_in[5];
    const float* bin_start = (const float*)d_in[6];

    float* a_scratch = (float*)d_ws;           // B_ * NV_ floats = 128 KB
    float* out       = (float*)d_out;          // [8, 512] fp32

    dim3 grid1(NV_ / 128, B_);
    hier_sim_bin_attn_kernel<<<grid1, 256, 0, stream>>>(
        doc_index, attn, emb, VvT, bin_diff, bin_start, a_scratch);

    hier_final_gemm_kernel<<<NODES_ / 16, 32, 0, stream>>>(
        a_scratch, phi, out);
}